// AdvancedGNNEncoder_46514495816137
// MI455X (gfx1250) — compile-verified
//
#include <hip/hip_runtime.h>
#include <hip/hip_bf16.h>
#include <math.h>

// ---------------------------------------------------------------------------
// GATv2 encoder (2 layers, H=4 heads) for MI455X / gfx1250 (wave32, WMMA).
// Dense transforms use V_WMMA_F32_16X16X4_F32 (fp32-exact vs reference).
// Sparse attention phase uses wave-per-edge kernels with shfl reductions and
// float atomics for segment max / sum / scatter-add.
// ---------------------------------------------------------------------------

typedef __attribute__((ext_vector_type(2))) float v2f;
typedef __attribute__((ext_vector_type(8))) float v8f;

#define WAVE 32

// ---------------------------------------------------------------------------
// Generic fill
// ---------------------------------------------------------------------------
__global__ void fill_kernel(float* __restrict__ p, float v, long long n) {
    long long i = (long long)blockIdx.x * blockDim.x + threadIdx.x;
    if (i < n) p[i] = v;
}

// ---------------------------------------------------------------------------
// Self-loop weight = mean of incoming edge weights per node (PyG fill='mean')
// ---------------------------------------------------------------------------
__global__ void selfloop_accum_kernel(const int* __restrict__ dst,
                                      const float* __restrict__ ew,
                                      float* __restrict__ sumw,
                                      float* __restrict__ cntw, int E) {
    int e = blockIdx.x * blockDim.x + threadIdx.x;
    if (e >= E) return;
    int d = dst[e];
    atomicAdd(&sumw[d], ew[e]);
    atomicAdd(&cntw[d], 1.0f);
}

__global__ void selfloop_w_kernel(const float* __restrict__ sumw,
                                  const float* __restrict__ cntw,
                                  float* __restrict__ loopw, int n) {
    int i = blockIdx.x * blockDim.x + threadIdx.x;
    if (i >= n) return;
    loopw[i] = sumw[i] / fmaxf(cntw[i], 1.0f);
}

// ---------------------------------------------------------------------------
// WMMA GEMM:  Y[rows x Mout] = X[rows x K] @ W[K x Mout] + bias
// One wave computes one 16x16 tile with V_WMMA_F32_16X16X4_F32.
//   A (16x4 f32):  lane l: m = l&15, kbase = (l>>4)*2 -> {A[m][kb], A[m][kb+1]}
//   B (4x16 f32):  lane l: n = l&15, kbase = (l>>4)*2 -> {B[kb][n], B[kb+1][n]}
//   C/D (16x16 f32): vgpr j at lane l -> row j + (l>>4)*8, col l&15
// rows (=N=50000) is a multiple of 16; K in {128}; Mout in {64,128,256}.
// ---------------------------------------------------------------------------
__global__ void gemm_bias_wmma_kernel(const float* __restrict__ X,
                                      const float* __restrict__ W,
                                      const float* __restrict__ bias,
                                      float* __restrict__ Y,
                                      int rows, int K, int Mout) {
    int gtid = blockIdx.x * blockDim.x + threadIdx.x;
    int wave = gtid >> 5;
    int lane = gtid & 31;
    int tilesN = Mout >> 4;
    int rowT = wave / tilesN;
    int colT = wave - rowT * tilesN;
    if (rowT * 16 >= rows) return;

    const int m    = lane & 15;
    const int half = lane >> 4;

    const float* __restrict__ Arow = X + (size_t)(rowT * 16 + m) * K;
    const float* __restrict__ Bcol = W + (size_t)(colT * 16 + m);

    v8f acc = {};
    for (int k0 = 0; k0 < K; k0 += 4) {
        const int ka = k0 + half * 2;
        v2f a;
        a.x = Arow[ka];
        a.y = Arow[ka + 1];
        v2f b;
        b.x = Bcol[(size_t)ka * Mout];
        b.y = Bcol[(size_t)(ka + 1) * Mout];
        acc = __builtin_amdgcn_wmma_f32_16x16x4_f32(
            /*neg_a=*/false, a, /*neg_b=*/false, b,
            /*c_mod=*/(short)0, acc, /*reuse_a=*/false, /*reuse_b=*/false);
    }

    const int col = colT * 16 + m;
    const float bv = bias[col];
#pragma unroll
    for (int j = 0; j < 8; ++j) {
        const int row = rowT * 16 + j + half * 8;
        Y[(size_t)row * Mout + col] = acc[j] + bv;
    }
}

// ---------------------------------------------------------------------------
// float atomic max via int/uint punning (valid with -inf initialization)
// ---------------------------------------------------------------------------
__device__ __forceinline__ void atomicMaxF(float* addr, float value) {
    if (value >= 0.0f)
        atomicMax((int*)addr, __float_as_int(value));
    else
        atomicMin((unsigned int*)addr, __float_as_uint(value));
}

// ---------------------------------------------------------------------------
// Edge pass 1: per-edge logits + segment max over dst.
// One wave per edge; lanes split C channels; 4 heads in registers.
// Virtual edges [E, E+N) are the self-loops with weight loopw[n].
// ---------------------------------------------------------------------------
template <int C>
__global__ void edge_logits_kernel(const int* __restrict__ src,
                                   const int* __restrict__ dst,
                                   const float* __restrict__ ew,
                                   const float* __restrict__ loopw,
                                   int E, int Etot,
                                   const float* __restrict__ xl,
                                   const float* __restrict__ xr,
                                   const float* __restrict__ We,
                                   const float* __restrict__ att,
                                   float* __restrict__ logits,
                                   float* __restrict__ maxb) {
    const int gtid = blockIdx.x * blockDim.x + threadIdx.x;
    const int e    = gtid >> 5;
    const int lane = gtid & 31;
    if (e >= Etot) return;

    int s, d;
    float w;
    if (e < E) { s = src[e]; d = dst[e]; w = ew[e]; }
    else       { s = d = e - E; w = loopw[s]; }

    const int HC = 4 * C;
    float lg0 = 0.f, lg1 = 0.f, lg2 = 0.f, lg3 = 0.f;

    for (int c = lane; c < C; c += WAVE) {
        const size_t sb = (size_t)s * HC;
        const size_t db = (size_t)d * HC;
#pragma unroll
        for (int h = 0; h < 4; ++h) {
            const int idx = h * C + c;
            float v = xl[sb + idx] + xr[db + idx] + w * We[idx];
            v = (v > 0.f) ? v : 0.2f * v;       // leaky_relu(., 0.2)
            const float t = v * att[idx];
            if (h == 0) lg0 += t; else if (h == 1) lg1 += t;
            else if (h == 2) lg2 += t; else lg3 += t;
        }
    }
#pragma unroll
    for (int off = 16; off > 0; off >>= 1) {
        lg0 += __shfl_xor(lg0, off, WAVE);
        lg1 += __shfl_xor(lg1, off, WAVE);
        lg2 += __shfl_xor(lg2, off, WAVE);
        lg3 += __shfl_xor(lg3, off, WAVE);
    }
    if (lane == 0) {
        float lg[4] = {lg0, lg1, lg2, lg3};
#pragma unroll
        for (int h = 0; h < 4; ++h) {
            logits[(size_t)e * 4 + h] = lg[h];
            atomicMaxF(&maxb[(size_t)d * 4 + h], lg[h]);
        }
    }
}

// ---------------------------------------------------------------------------
// Edge pass 2: p = exp(logit - max[dst]); segment-sum denom. In-place on logits.
// ---------------------------------------------------------------------------
__global__ void edge_softmax_kernel(const int* __restrict__ dst, int E, int Etot,
                                    float* __restrict__ logits,
                                    const float* __restrict__ maxb,
                                    float* __restrict__ denb) {
    const int e = blockIdx.x * blockDim.x + threadIdx.x;
    if (e >= Etot) return;
    const int d = (e < E) ? dst[e] : (e - E);
#pragma unroll
    for (int h = 0; h < 4; ++h) {
        const float p = __expf(logits[(size_t)e * 4 + h] - maxb[(size_t)d * 4 + h]);
        logits[(size_t)e * 4 + h] = p;
        atomicAdd(&denb[(size_t)d * 4 + h], p);
    }
}

// ---------------------------------------------------------------------------
// Edge pass 3: scatter out[dst] += alpha * xl[src].  Wave per edge.
// ---------------------------------------------------------------------------
template <int C>
__global__ void edge_aggregate_kernel(const int* __restrict__ src,
                                      const int* __restrict__ dst,
                                      int E, int Etot,
                                      const float* __restrict__ pnum,
                                      const float* __restrict__ denb,
                                      const float* __restrict__ xl,
                                      float* __restrict__ outacc) {
    const int gtid = blockIdx.x * blockDim.x + threadIdx.x;
    const int e    = gtid >> 5;
    const int lane = gtid & 31;
    if (e >= Etot) return;

    int s, d;
    if (e < E) { s = src[e]; d = dst[e]; }
    else       { s = d = e - E; }

    float alpha[4];
#pragma unroll
    for (int h = 0; h < 4; ++h)
        alpha[h] = pnum[(size_t)e * 4 + h] / denb[(size_t)d * 4 + h];

    const int HC = 4 * C;
    const size_t sb = (size_t)s * HC;
    const size_t db = (size_t)d * HC;
    for (int c = lane; c < C; c += WAVE) {
#pragma unroll
        for (int h = 0; h < 4; ++h) {
            const int idx = h * C + c;
            atomicAdd(&outacc[db + idx], xl[sb + idx] * alpha[h]);
        }
    }
}

// ---------------------------------------------------------------------------
// Finalize layer 0 (concat heads, D=128): elu -> LayerNorm -> *g+be -> +res
// One wave per node; lane handles channels lane, lane+32, lane+64, lane+96.
// ---------------------------------------------------------------------------
__global__ void finalize_concat_kernel(const float* __restrict__ acc,
                                       const float* __restrict__ bias,
                                       const float* __restrict__ g,
                                       const float* __restrict__ be,
                                       const float* __restrict__ res,
                                       float* __restrict__ out, int nN) {
    const int gtid = blockIdx.x * blockDim.x + threadIdx.x;
    const int node = gtid >> 5;
    const int lane = gtid & 31;
    if (node >= nN) return;
    const int D = 128;

    float v[4], s = 0.f, s2 = 0.f;
#pragma unroll
    for (int i = 0; i < 4; ++i) {
        const int ch = lane + i * WAVE;
        float x = acc[(size_t)node * D + ch] + bias[ch];
        x = (x > 0.f) ? x : (__expf(x) - 1.f);   // elu
        v[i] = x;
        s += x;
        s2 += x * x;
    }
#pragma unroll
    for (int off = 16; off > 0; off >>= 1) {
        s  += __shfl_xor(s, off, WAVE);
        s2 += __shfl_xor(s2, off, WAVE);
    }
    const float mu  = s / (float)D;
    const float var = s2 / (float)D - mu * mu;
    const float inv = rsqrtf(var + 1e-5f);
#pragma unroll
    for (int i = 0; i < 4; ++i) {
        const int ch = lane + i * WAVE;
        out[(size_t)node * D + ch] =
            (v[i] - mu) * inv * g[ch] + be[ch] + res[(size_t)node * D + ch];
    }
}

// ---------------------------------------------------------------------------
// Finalize layer 1 (mean over 4 heads, D=64): mean -> +b -> elu -> LN -> +res
// ---------------------------------------------------------------------------
__global__ void finalize_mean_kernel(const float* __restrict__ acc,   // [n][h*64+c]
                                     const float* __restrict__ bias,
                                     const float* __restrict__ g,
                                     const float* __restrict__ be,
                                     const float* __restrict__ res,   // [n][64]
                                     float* __restrict__ out, int nN) {
    const int gtid = blockIdx.x * blockDim.x + threadIdx.x;
    const int node = gtid >> 5;
    const int lane = gtid & 31;
    if (node >= nN) return;
    const int D = 64;

    float v[2], s = 0.f, s2 = 0.f;
#pragma unroll
    for (int i = 0; i < 2; ++i) {
        const int c = lane + i * WAVE;
        const size_t nb = (size_t)node * 256;
        float x = 0.25f * (acc[nb + 0 * D + c] + acc[nb + 1 * D + c] +
                           acc[nb + 2 * D + c] + acc[nb + 3 * D + c]) + bias[c];
        x = (x > 0.f) ? x : (__expf(x) - 1.f);   // elu
        v[i] = x;
        s += x;
        s2 += x * x;
    }
#pragma unroll
    for (int off = 16; off > 0; off >>= 1) {
        s  += __shfl_xor(s, off, WAVE);
        s2 += __shfl_xor(s2, off, WAVE);
    }
    const float mu  = s / (float)D;
    const float var = s2 / (float)D - mu * mu;
    const float inv = rsqrtf(var + 1e-5f);
#pragma unroll
    for (int i = 0; i < 2; ++i) {
        const int c = lane + i * WAVE;
        out[(size_t)node * D + c] =
            (v[i] - mu) * inv * g[c] + be[c] + res[(size_t)node * D + c];
    }
}

// ---------------------------------------------------------------------------
// Host-side orchestration
// ---------------------------------------------------------------------------
static inline int blocks_for(long long work, int bs) {
    return (int)((work + bs - 1) / bs);
}

extern "C" void kernel_launch(void* const* d_in, const int* in_sizes, int n_in,
                              void* d_out, int out_size, void* d_ws, size_t ws_size,
                              hipStream_t stream) {
    const float* x    = (const float*)d_in[0];
    const int*   ei   = (const int*)d_in[1];
    const float* ew   = (const float*)d_in[2];
    const float* Wl0  = (const float*)d_in[3];
    const float* bl0  = (const float*)d_in[4];
    const float* Wr0  = (const float*)d_in[5];
    const float* br0  = (const float*)d_in[6];
    const float* We0  = (const float*)d_in[7];
    const float* att0 = (const float*)d_in[8];
    const float* b0   = (const float*)d_in[9];
    const float* g0   = (const float*)d_in[10];
    const float* be0  = (const float*)d_in[11];
    const float* Wl1  = (const float*)d_in[12];
    const float* bl1  = (const float*)d_in[13];
    const float* Wr1  = (const float*)d_in[14];
    const float* br1  = (const float*)d_in[15];
    const float* We1  = (const float*)d_in[16];
    const float* att1 = (const float*)d_in[17];
    const float* b1   = (const float*)d_in[18];
    const float* g1   = (const float*)d_in[19];
    const float* be1  = (const float*)d_in[20];
    const float* P0   = (const float*)d_in[21];
    const float* pb0  = (const float*)d_in[22];
    const float* P1   = (const float*)d_in[23];
    const float* pb1  = (const float*)d_in[24];

    const int IN  = 128, HIDD = 128, O1 = 256, OUT = 64;
    const int N = in_sizes[0] / IN;
    const int E = in_sizes[2];
    const int Etot = E + N;           // real edges + self loops
    const int* srcp = ei;             // edge_index row 0
    const int* dstp = ei + E;         // edge_index row 1

    // ---- workspace carve (floats) ----
    float* ws = (float*)d_ws;
    size_t off = 0;
    float* sumw  = ws + off; off += (size_t)N;
    float* cntw  = ws + off; off += (size_t)N;
    float* loopw = ws + off; off += (size_t)N;
    float* XL    = ws + off; off += (size_t)N * O1;    // holds N*128 (L0) / N*256 (L1)
    float* XR    = ws + off; off += (size_t)N * O1;
    float* ACC   = ws + off; off += (size_t)N * O1;
    float* RES   = ws + off; off += (size_t)N * HIDD;  // residual proj (128 / 64 used)
    float* HCUR  = ws + off; off += (size_t)N * HIDD;  // layer-0 output features
    float* MAXB  = ws + off; off += (size_t)N * 4;
    float* DENB  = ws + off; off += (size_t)N * 4;
    float* LOGIT = ws + off; off += (size_t)Etot * 4;
    (void)ws_size; (void)n_in; (void)out_size;

    const int BS = 256;
    const float NEG_INF = -__builtin_inff();

    // ---- self-loop weights (mean of incoming edge weights) ----
    fill_kernel<<<blocks_for(2LL * N, BS), BS, 0, stream>>>(sumw, 0.f, 2LL * N); // sumw+cntw contiguous
    selfloop_accum_kernel<<<blocks_for(E, BS), BS, 0, stream>>>(dstp, ew, sumw, cntw, E);
    selfloop_w_kernel<<<blocks_for(N, BS), BS, 0, stream>>>(sumw, cntw, loopw, N);

    // =============================== Layer 0 ===============================
    {
        const long long t128 = (long long)(N / 16) * (HIDD / 16) * WAVE;
        gemm_bias_wmma_kernel<<<blocks_for(t128, BS), BS, 0, stream>>>(x, Wl0, bl0, XL, N, IN, HIDD);
        gemm_bias_wmma_kernel<<<blocks_for(t128, BS), BS, 0, stream>>>(x, Wr0, br0, XR, N, IN, HIDD);
        gemm_bias_wmma_kernel<<<blocks_for(t128, BS), BS, 0, stream>>>(x, P0, pb0, RES, N, IN, HIDD);

        fill_kernel<<<blocks_for(4LL * N, BS), BS, 0, stream>>>(MAXB, NEG_INF, 4LL * N);
        fill_kernel<<<blocks_for(4LL * N, BS), BS, 0, stream>>>(DENB, 0.f, 4LL * N);
        fill_kernel<<<blocks_for((long long)N * HIDD, BS), BS, 0, stream>>>(ACC, 0.f, (long long)N * HIDD);

        const long long ethreads = (long long)Etot * WAVE;
        edge_logits_kernel<32><<<blocks_for(ethreads, BS), BS, 0, stream>>>(
            srcp, dstp, ew, loopw, E, Etot, XL, XR, We0, att0, LOGIT, MAXB);
        edge_softmax_kernel<<<blocks_for(Etot, BS), BS, 0, stream>>>(
            dstp, E, Etot, LOGIT, MAXB, DENB);
        edge_aggregate_kernel<32><<<blocks_for(ethreads, BS), BS, 0, stream>>>(
            srcp, dstp, E, Etot, LOGIT, DENB, XL, ACC);

        finalize_concat_kernel<<<blocks_for((long long)N * WAVE, BS), BS, 0, stream>>>(
            ACC, b0, g0, be0, RES, HCUR, N);
    }

    // =============================== Layer 1 ===============================
    {
        const long long t256 = (long long)(N / 16) * (O1 / 16) * WAVE;
        const long long t64  = (long long)(N / 16) * (OUT / 16) * WAVE;
        gemm_bias_wmma_kernel<<<blocks_for(t256, BS), BS, 0, stream>>>(HCUR, Wl1, bl1, XL, N, HIDD, O1);
        gemm_bias_wmma_kernel<<<blocks_for(t256, BS), BS, 0, stream>>>(HCUR, Wr1, br1, XR, N, HIDD, O1);
        gemm_bias_wmma_kernel<<<blocks_for(t64, BS), BS, 0, stream>>>(HCUR, P1, pb1, RES, N, HIDD, OUT);

        fill_kernel<<<blocks_for(4LL * N, BS), BS, 0, stream>>>(MAXB, NEG_INF, 4LL * N);
        fill_kernel<<<blocks_for(4LL * N, BS), BS, 0, stream>>>(DENB, 0.f, 4LL * N);
        fill_kernel<<<blocks_for((long long)N * O1, BS), BS, 0, stream>>>(ACC, 0.f, (long long)N * O1);

        const long long ethreads = (long long)Etot * WAVE;
        edge_logits_kernel<64><<<blocks_for(ethreads, BS), BS, 0, stream>>>(
            srcp, dstp, ew, loopw, E, Etot, XL, XR, We1, att1, LOGIT, MAXB);
        edge_softmax_kernel<<<blocks_for(Etot, BS), BS, 0, stream>>>(
            dstp, E, Etot, LOGIT, MAXB, DENB);
        edge_aggregate_kernel<64><<<blocks_for(ethreads, BS), BS, 0, stream>>>(
            srcp, dstp, E, Etot, LOGIT, DENB, XL, ACC);

        finalize_mean_kernel<<<blocks_for((long long)N * WAVE, BS), BS, 0, stream>>>(
            ACC, b1, g1, be1, RES, (float*)d_out, N);
    }
}